// EMARecurrent_84293028151613
// MI455X (gfx1250) — compile-verified
//
#include <hip/hip_runtime.h>
#include <cstdint>

typedef float  v2f __attribute__((ext_vector_type(2)));
typedef float  v4f __attribute__((ext_vector_type(4)));
typedef float  v8f __attribute__((ext_vector_type(8)));
typedef unsigned int v4u __attribute__((ext_vector_type(4)));
typedef int    v4i __attribute__((ext_vector_type(4)));
typedef int    v8i __attribute__((ext_vector_type(8)));

#define EMA_A 0.4f
#define EMA_C 0.6f

constexpr int kT        = 4096;   // timesteps per row
constexpr int kRows     = 8192;   // B*D
constexpr int kSeg      = 512;    // timesteps per independent segment
constexpr int kChunk    = 64;     // timesteps per TDM tile
constexpr int kInStride = 66;     // dwords/row in TDM-padded input LDS tile (64 + 2 pad)
constexpr int kOutStride= 68;     // dwords/row in output transpose LDS tile
constexpr int kWaves    = 4;      // waves per workgroup

__device__ __forceinline__ float cpowi(int e) {
  float r = 1.0f;
  for (int i = 0; i < e; ++i) r *= EMA_C;
  return r;
}

// Issue a TDM load of a 16-row x 64-timestep fp32 tile into LDS.
// 2-D descriptor: dim0 = time (stride 1), dim1 = row (stride kT elements).
// LDS padding: +2 dwords every 64 dwords -> row stride 66 dwords (bank-conflict-free
// transposed reads: bank = (2*row + t) % 64).
__device__ __forceinline__ void tdm_load_tile(unsigned lds_byte, const float* gptr) {
  unsigned long long ga = (unsigned long long)(uintptr_t)gptr;
  v4u g0;
  g0[0] = 1u;                                                  // count=1, gather off
  g0[1] = lds_byte;                                            // LDS dest (bytes)
  g0[2] = (unsigned)(ga & 0xFFFFFFFFull);                      // global addr [31:0]
  g0[3] = (unsigned)((ga >> 32) & 0x1FFFFFFull) | (2u << 30);  // addr [56:32] | type=2

  const unsigned td0 = 1u << 20;   // tensor_dim0 (large: no OOB clipping needed)
  const unsigned td1 = 1u << 20;   // tensor_dim1
  const unsigned st0 = (unsigned)kT; // dim0 stride = row stride in elements
  v8i g1;
  g1[0] = (int)((2u << 16)     // data_size = 4 bytes
              | (1u << 20)     // pad_enable
              | (5u << 22)     // pad_interval code 5 -> every 64 dwords
              | (1u << 25));   // pad_amount  code 1 -> 2 dwords
  g1[1] = (int)((td0 & 0xFFFFu) << 16);                      // tensor_dim0 lo16
  g1[2] = (int)((td0 >> 16) | ((td1 & 0xFFFFu) << 16));      // dim0 hi | dim1 lo16
  g1[3] = (int)((td1 >> 16) | ((unsigned)kChunk << 16));     // dim1 hi | tile_dim0=64
  g1[4] = 16;                                                // tile_dim1=16 rows, tile_dim2=0
  g1[5] = (int)st0;                                          // tensor_dim0_stride lo32
  g1[6] = 0;                                                 // stride0 hi | stride1 lo (unused, 2D)
  g1[7] = 0;
  v4i z4 = {0, 0, 0, 0};
  v8i z8 = {0, 0, 0, 0, 0, 0, 0, 0};
  __builtin_amdgcn_tensor_load_to_lds(g0, g1, z4, z4, z8, 0);
}

extern "C" __global__ __launch_bounds__(128)
void EMARecurrent_wmma_scan_kernel(const float* __restrict__ inp,
                                   const float* __restrict__ hid,
                                   float* __restrict__ out) {
  __shared__ float ldsIn[kWaves][2][16 * kInStride];   // double-buffered TDM dest
  __shared__ float ldsOut[kWaves][16 * kOutStride];    // output transpose buffer

  const int wave = threadIdx.x >> 5;
  const int lane = threadIdx.x & 31;
  const int n    = lane & 15;     // matrix column (row of data tile)
  const int hi   = lane >> 4;     // lane half

  const int unit = blockIdx.x * kWaves + wave;   // 0..4095
  const int rt   = unit >> 3;                    // row tile 0..511
  const int seg  = unit & 7;                     // time segment 0..7
  const int row0 = rt << 4;

  // ---- per-lane constants -------------------------------------------------
  // A operand (16x4 f32 tile of M[t][k] = a*c^(t-k), t>=k), ISA layout:
  // vgpr j, lanes 0-15: k_local=j ; lanes 16-31: k_local=j+2 ; t = lane%16.
  v2f Am[4];
#pragma unroll
  for (int kc = 0; kc < 4; ++kc)
#pragma unroll
    for (int j = 0; j < 2; ++j) {
      int k = kc * 4 + j + 2 * hi;
      Am[kc][j] = (n >= k) ? EMA_A * cpowi(n - k) : 0.0f;
    }
  // carry-in scale for C[t][n] = c^(t+1) * h[n], t = j + 8*hi
  float cpw[8];
#pragma unroll
  for (int j = 0; j < 8; ++j) cpw[j] = cpowi(j + 1 + 8 * hi);

  // ---- segment setup ------------------------------------------------------
  const int warm   = (seg != 0) ? 1 : 0;      // one 64-step warm-up chunk
  const int nch    = 8 + warm;                // chunks of 64 timesteps
  const int tstart = seg * kSeg - warm * kChunk;

  float* bin0 = &ldsIn[wave][0][0];
  float* bin1 = &ldsIn[wave][1][0];
  float* bout = &ldsOut[wave][0];
  const unsigned lo0 = (unsigned)(size_t)bin0;
  const unsigned lo1 = (unsigned)(size_t)bin1;

  const float* gbase = inp + (size_t)row0 * kT + tstart;

  // pre-issue the first two TDM tiles
  tdm_load_tile(lo0, gbase);
  if (nch > 1) tdm_load_tile(lo1, gbase + kChunk);

  // carry (already replicated per lane: each lane holds h for its n)
  float hrow = (seg == 0) ? hid[row0 + n] : 0.0f;

  for (int i = 0; i < nch; ++i) {
    if (i + 1 < nch) __builtin_amdgcn_s_wait_tensorcnt(1);  // tile i landed
    else             __builtin_amdgcn_s_wait_tensorcnt(0);
    asm volatile("" ::: "memory");

    const float* bin = (i & 1) ? bin1 : bin0;

    // ---- 4 blocks of 16 timesteps, each = 4 chained WMMA f32 16x16x4 ----
#pragma unroll
    for (int blk = 0; blk < 4; ++blk) {
      v8f acc;
#pragma unroll
      for (int j = 0; j < 8; ++j) acc[j] = cpw[j] * hrow;   // C = c^(t+1)*h_prev
#pragma unroll
      for (int kc = 0; kc < 4; ++kc) {
        // B operand: vgpr j holds x[n][t0 + kc*4 + j + 2*hi] -> one b64 LDS read
        const int toff = blk * 16 + kc * 4 + 2 * hi;
        v2f b = *(const v2f*)&bin[n * kInStride + toff];
        acc = __builtin_amdgcn_wmma_f32_16x16x4_f32(
            false, Am[kc], false, b, (short)0, acc, false, false);
      }
      // D[t][n] -> LDS transpose buffer (row n, t = blk*16 + j + 8*hi)
#pragma unroll
      for (int j = 0; j < 8; ++j)
        bout[n * kOutStride + blk * 16 + 8 * hi + j] = acc[j];
      // new carry h[n] = y[15][n] lives in acc[7] of lane 16+n
      hrow = __int_as_float(
          __builtin_amdgcn_ds_bpermute((n | 16) << 2, __float_as_int(acc[7])));
    }

    // ---- coalesced store of the 16x64 tile (skip warm-up chunk) ----
    if (!(warm && i == 0)) {
      const int tchunk = tstart + i * kChunk;
#pragma unroll
      for (int g = 0; g < 8; ++g) {
        const int rl = 2 * g + hi;   // lanes 0-15: row 2g, lanes 16-31: row 2g+1
        v4f v = *(const v4f*)&bout[rl * kOutStride + (n << 2)];
        *(v4f*)&out[(size_t)(row0 + rl) * kT + tchunk + (n << 2)] = v;
      }
    }

    // all LDS reads of buffer (i&1) done -> safe to refill it
    asm volatile("s_wait_dscnt 0" ::: "memory");
    if (i + 2 < nch)
      tdm_load_tile((i & 1) ? lo1 : lo0, gbase + (size_t)(i + 2) * kChunk);
  }
  __builtin_amdgcn_s_wait_tensorcnt(0);
}

extern "C" void kernel_launch(void* const* d_in, const int* in_sizes, int n_in,
                              void* d_out, int out_size, void* d_ws, size_t ws_size,
                              hipStream_t stream) {
  (void)in_sizes; (void)n_in; (void)out_size; (void)d_ws; (void)ws_size;
  const float* inp = (const float*)d_in[0];   // [8,1024,4096] f32
  const float* hid = (const float*)d_in[1];   // [8,1024,1]   f32
  float* out = (float*)d_out;                 // [8,1024,4096] f32
  // 8192 rows / 16 = 512 row tiles; 8 time segments; 4 waves per block
  const int units  = (kRows / 16) * (kT / kSeg);   // 4096 waves
  const int blocks = units / kWaves;               // 1024
  EMARecurrent_wmma_scan_kernel<<<blocks, 128, 0, stream>>>(inp, hid, out);
}